// LGCN_Encoder_35003983462545
// MI455X (gfx1250) — compile-verified
//
#include <hip/hip_runtime.h>

// ---------------------------------------------------------------------------
// LightGCN spectral propagation for MI455X (gfx1250), f32 WMMA path.
//   ego  : [N_TOTAL, 64]  (concat of user/item embeddings)
//   v    : [N_TOTAL, 512]
//   L layers of: proj = v^T @ ego ; ego = v @ (filters[k] ⊙ proj) ; acc += ego
//   out  = acc / (L+1)
//
// GEMM inner loops: branchless (zero-filled LDS operands + clamped addresses),
// no compiler unroll (stable in-place accumulator bank), 32x64 tile per wave
// so every B fetch feeds two v_wmma_f32_16x16x4_f32.
// ---------------------------------------------------------------------------

typedef __attribute__((ext_vector_type(2))) float v2f;
typedef __attribute__((ext_vector_type(4))) float v4f;
typedef __attribute__((ext_vector_type(8))) float v8f;

#define N_USERS 29858
#define N_ITEMS 40981
#define N_TOTAL (N_USERS + N_ITEMS)    // 70839
#define DIM     64
#define FDIM    512
#define N_LAYERS 3
#define NCHUNKS 128
#define CHUNK   556                    // multiple of 4; 128*556 = 71168 >= 70839
#define QUADS   (CHUNK / 4)            // 139

static_assert(NCHUNKS * CHUNK >= N_TOTAL, "chunk coverage");

#define WMMA_F32(A, B, C) \
    __builtin_amdgcn_wmma_f32_16x16x4_f32(false, (A), false, (B), (short)0, (C), false, false)

// ---------------------------------------------------------------------------
// ego = concat(user, item); acc = ego
// ---------------------------------------------------------------------------
__global__ void lgcn_init(const float* __restrict__ user,
                          const float* __restrict__ item,
                          float* __restrict__ ego,
                          float* __restrict__ acc) {
    int i = blockIdx.x * blockDim.x + threadIdx.x;
    if (i < N_TOTAL * DIM) {
        float val = (i < N_USERS * DIM) ? user[i] : item[i - N_USERS * DIM];
        ego[i] = val;
        acc[i] = val;
    }
}

// ---------------------------------------------------------------------------
// Stage A: partial[chunk] = v[chunk,:]^T @ ego[chunk,:]   (split-K, no atomics)
//
// Block = 4 waves; each wave owns a 32(f) x 64(d) tile; gridDim.y = 4 covers
// all 512 f-cols. ego chunk staged in LDS as row pairs (zero-filled past the
// valid rows) so the K loop is fully branchless.
// Per quad: 4x global_load_b32 (A) + 2x ds_load_2addr_b64 (B) + 8x WMMA.
// ---------------------------------------------------------------------------
__global__ __launch_bounds__(128)
void lgcn_projT(const float* __restrict__ v,
                const float* __restrict__ ego,
                float* __restrict__ partial /* [NCHUNKS][FDIM][DIM] */) {
    __shared__ v2f lds_ego[(CHUNK / 2) * DIM];   // 278*64*8B = 139 KB

    const int chunk   = blockIdx.x;
    const int n_begin = chunk * CHUNK;
    const int n_valid = min(N_TOTAL - n_begin, CHUNK);

    // Stage ego chunk (branchless: clamped address + select).
    for (int idx = threadIdx.x; idx < (CHUNK / 2) * DIM; idx += blockDim.x) {
        const int j = idx >> 6;
        const int d = idx & 63;
        const int r0 = 2 * j, r1 = 2 * j + 1;
        const float e0 = ego[(size_t)(n_begin + min(r0, n_valid - 1)) * DIM + d];
        const float e1 = ego[(size_t)(n_begin + min(r1, n_valid - 1)) * DIM + d];
        v2f val;
        val.x = (r0 < n_valid) ? e0 : 0.0f;
        val.y = (r1 < n_valid) ? e1 : 0.0f;
        lds_ego[idx] = val;
    }
    __syncthreads();

    const int lane  = threadIdx.x & 31;
    const int wave  = threadIdx.x >> 5;
    const int f0    = (blockIdx.y * 4 + wave) * 32;   // 32 f-cols per wave
    const int row   = lane & 15;
    const int half  = lane >> 4;         // 0 or 1
    const int khalf = half * 2;

    v8f c0[4] = {{}, {}, {}, {}};        // f-cols [f0,    f0+16)
    v8f c1[4] = {{}, {}, {}, {}};        // f-cols [f0+16, f0+32)

    if (n_valid == CHUNK) {
        // Fast path: fully in-bounds, affine pointer walk.
        const float* ap = v + (size_t)(n_begin + khalf) * FDIM + f0 + row;
#pragma unroll 1
        for (int q = 0; q < QUADS; ++q) {
            v2f a0, a1;
            a0.x = ap[0];         a0.y = ap[FDIM];
            a1.x = ap[16];        a1.y = ap[FDIM + 16];
            const v2f* bp = &lds_ego[(q * 2 + half) * DIM + row];
#pragma unroll
            for (int t = 0; t < 4; ++t) {
                v2f b = bp[t * 16];
                c0[t] = WMMA_F32(a0, b, c0[t]);
                c1[t] = WMMA_F32(a1, b, c1[t]);
            }
            ap += 4 * FDIM;
        }
    } else {
        // Ragged chunk: clamp A rows; B is zero there, so products vanish.
#pragma unroll 1
        for (int q = 0; q < QUADS; ++q) {
            const int na  = n_begin + q * 4 + khalf;
            const size_t ra0 = (size_t)min(na,     N_TOTAL - 1) * FDIM;
            const size_t ra1 = (size_t)min(na + 1, N_TOTAL - 1) * FDIM;
            v2f a0, a1;
            a0.x = v[ra0 + f0 + row];        a0.y = v[ra1 + f0 + row];
            a1.x = v[ra0 + f0 + 16 + row];   a1.y = v[ra1 + f0 + 16 + row];
            const v2f* bp = &lds_ego[(q * 2 + half) * DIM + row];
#pragma unroll
            for (int t = 0; t < 4; ++t) {
                v2f b = bp[t * 16];
                c0[t] = WMMA_F32(a0, b, c0[t]);
                c1[t] = WMMA_F32(a1, b, c1[t]);
            }
        }
    }

    // C layout: VGPR j -> M = j + 8*half; N = (lane&15) + 16*t
    float* outp = partial + (size_t)chunk * FDIM * DIM;
    const int mbase = f0 + (half << 3);
#pragma unroll
    for (int t = 0; t < 4; ++t) {
#pragma unroll
        for (int j = 0; j < 8; ++j) {
            outp[(size_t)(mbase + j)      * DIM + t * 16 + row] = c0[t][j];
            outp[(size_t)(mbase + 16 + j) * DIM + t * 16 + row] = c1[t][j];
        }
    }
}

// ---------------------------------------------------------------------------
// proj[f,d] = filters[layer][f] * sum_c partial[c][f][d]
// ---------------------------------------------------------------------------
__global__ void lgcn_reduce(const float* __restrict__ partial,
                            const float* __restrict__ filt /* [FDIM] */,
                            float* __restrict__ proj /* [FDIM][DIM] */) {
    int i = blockIdx.x * blockDim.x + threadIdx.x;
    if (i < FDIM * DIM) {
        float s = 0.0f;
        for (int c = 0; c < NCHUNKS; ++c)
            s += partial[(size_t)c * FDIM * DIM + i];
        proj[i] = s * filt[i / DIM];
    }
}

// ---------------------------------------------------------------------------
// Stage B: ego = v @ proj ; acc += ego
// proj staged in LDS as row pairs (128 KB -> 2 blocks/WGP of the 320 KB pool).
// 8 waves/block, 32 output rows per wave, K loop branchless (clamped v rows).
// Per quad: 2x global_load_b64 (A) + 2x ds_load_2addr_b64 (B) + 8x WMMA.
// ---------------------------------------------------------------------------
__global__ __launch_bounds__(256)
void lgcn_prop(const float* __restrict__ v,
               const float* __restrict__ proj,
               float* __restrict__ ego,
               float* __restrict__ acc) {
    __shared__ v2f lds_proj[(FDIM / 2) * DIM];   // 256*64*8B = 128 KB

    for (int idx = threadIdx.x; idx < (FDIM / 2) * DIM; idx += blockDim.x) {
        const int j = idx >> 6;
        const int d = idx & 63;
        v2f val;
        val.x = proj[(size_t)(2 * j)     * DIM + d];
        val.y = proj[(size_t)(2 * j + 1) * DIM + d];
        lds_proj[idx] = val;
    }
    __syncthreads();

    const int lane  = threadIdx.x & 31;
    const int wave  = threadIdx.x >> 5;
    const int m0    = blockIdx.x * 256 + wave * 32;   // 32 output rows per wave
    const int row   = lane & 15;
    const int half  = lane >> 4;
    const int khalf = half * 2;

    const int m = m0 + row;
    const float* vrow0 = v + (size_t)min(m,      N_TOTAL - 1) * FDIM;
    const float* vrow1 = v + (size_t)min(m + 16, N_TOTAL - 1) * FDIM;

    v8f c0[4] = {{}, {}, {}, {}};        // rows [m0,    m0+16)
    v8f c1[4] = {{}, {}, {}, {}};        // rows [m0+16, m0+32)

#pragma unroll 1
    for (int k = 0; k < FDIM; k += 4) {
        v2f a0 = *(const v2f*)(vrow0 + k + khalf);   // global_load_b64
        v2f a1 = *(const v2f*)(vrow1 + k + khalf);
        if (k < FDIM - 64) {
            __builtin_prefetch(vrow0 + k + 64, 0, 1);
            __builtin_prefetch(vrow1 + k + 64, 0, 1);
        }
        const v2f* bp = &lds_proj[((k >> 1) + half) * DIM + row];
#pragma unroll
        for (int t = 0; t < 4; ++t) {
            v2f b = bp[t * 16];                      // ds_load (2addr_b64 pairs)
            c0[t] = WMMA_F32(a0, b, c0[t]);
            c1[t] = WMMA_F32(a1, b, c1[t]);
        }
    }

    const int mbase = m0 + (half << 3);
#pragma unroll
    for (int j = 0; j < 8; ++j) {
#pragma unroll
        for (int t = 0; t < 4; ++t) {
            const int r0 = mbase + j;
            const int r1 = mbase + 16 + j;
            if (r0 < N_TOTAL) {
                const size_t idx = (size_t)r0 * DIM + t * 16 + row;
                ego[idx] = c0[t][j];
                acc[idx] += c0[t][j];
            }
            if (r1 < N_TOTAL) {
                const size_t idx = (size_t)r1 * DIM + t * 16 + row;
                ego[idx] = c1[t][j];
                acc[idx] += c1[t][j];
            }
        }
    }
}

// ---------------------------------------------------------------------------
// out = acc / (N_LAYERS + 1)
// ---------------------------------------------------------------------------
__global__ void lgcn_final(const float* __restrict__ acc,
                           float* __restrict__ out) {
    int i = blockIdx.x * blockDim.x + threadIdx.x;
    if (i < N_TOTAL * DIM) out[i] = acc[i] * 0.25f;
}

// ---------------------------------------------------------------------------
extern "C" void kernel_launch(void* const* d_in, const int* in_sizes, int n_in,
                              void* d_out, int out_size, void* d_ws, size_t ws_size,
                              hipStream_t stream) {
    const float* user    = (const float*)d_in[0];
    const float* item    = (const float*)d_in[1];
    const float* v       = (const float*)d_in[2];
    const float* filters = (const float*)d_in[3];

    // Workspace layout (floats): ego | acc | partial | proj
    float* ego     = (float*)d_ws;
    float* acc     = ego + (size_t)N_TOTAL * DIM;
    float* partial = acc + (size_t)N_TOTAL * DIM;
    float* proj    = partial + (size_t)NCHUNKS * FDIM * DIM;

    const int totalND = N_TOTAL * DIM;
    lgcn_init<<<(totalND + 255) / 256, 256, 0, stream>>>(user, item, ego, acc);

    for (int layer = 0; layer < N_LAYERS; ++layer) {
        dim3 gA(NCHUNKS, 4);   // 128 chunks x (4 blocks * 4 waves * 32 f) = 512 f
        lgcn_projT<<<gA, 128, 0, stream>>>(v, ego, partial);

        lgcn_reduce<<<(FDIM * DIM + 255) / 256, 256, 0, stream>>>(
            partial, filters + (size_t)layer * FDIM, proj);

        const int nblocksB = (N_TOTAL + 255) / 256;
        lgcn_prop<<<nblocksB, 256, 0, stream>>>(v, proj, ego, acc);
    }

    lgcn_final<<<(totalND + 255) / 256, 256, 0, stream>>>(acc, (float*)d_out);
}